// MyGNN_22359599743272
// MI455X (gfx1250) — compile-verified
//
#include <hip/hip_runtime.h>

#define GN 50000
#define GE 800000
#define GD 128
#define NLAYER 4

typedef __attribute__((ext_vector_type(2))) float v2f;
typedef __attribute__((ext_vector_type(8))) float v8f;

// ---------------- degree / normalization ----------------

__global__ void k_zero(float* p, int n) {
    int i = blockIdx.x * blockDim.x + threadIdx.x;
    if (i < n) p[i] = 0.0f;
}

__global__ void k_deg(const int* __restrict__ dst, float* deg, int e) {
    int i = blockIdx.x * blockDim.x + threadIdx.x;
    if (i < e) atomicAdd(&deg[dst[i]], 1.0f);
}

__global__ void k_rsqrt(float* dis, int n) {
    int i = blockIdx.x * blockDim.x + threadIdx.x;
    if (i < n) dis[i] = rsqrtf(1.0f + dis[i]);
}

// ---------------- fp32 WMMA GEMM: out[n x 128] = A[n x 128] @ W[128 x 128] ----
// grid.x = n/16 (n multiple of 16), block = 256 (8 waves); wave j does column
// tile [16j, 16j+16). A tile staged in LDS (stride 132 to avoid bank conflicts).

__global__ void k_gemm128(const float* __restrict__ A,
                          const float* __restrict__ W,
                          float* __restrict__ out) {
    __shared__ float As[16 * 132];
    const int rowBase = blockIdx.x * 16;
    const int tid = threadIdx.x;

    for (int i = tid; i < 16 * 128; i += 256) {
        int r = i >> 7, c = i & 127;
        As[r * 132 + c] = A[(size_t)(rowBase + r) * GD + c];
    }
    __syncthreads();

    const int wave  = tid >> 5;          // 0..7 -> N tile
    const int lane  = tid & 31;
    const int m     = lane & 15;         // row within tile (A) / col within tile (B)
    const int khalf = (lane >> 4) * 2;   // 0 or 2
    const int n     = wave * 16 + m;

    v8f c = {};
    #pragma unroll
    for (int k = 0; k < 128; k += 4) {
        v2f a, b;
        a.x = As[m * 132 + k + khalf];
        a.y = As[m * 132 + k + khalf + 1];
        b.x = W[(size_t)(k + khalf) * GD + n];
        b.y = W[(size_t)(k + khalf + 1) * GD + n];
        c = __builtin_amdgcn_wmma_f32_16x16x4_f32(
                false, a, false, b, (short)0, c, false, false);
    }

    const int rOff = (lane >> 4) * 8;    // C layout: VGPR i -> M = i (+8 hi half)
    #pragma unroll
    for (int i = 0; i < 8; ++i)
        out[(size_t)(rowBase + rOff + i) * GD + n] = c[i];
}

// ---------------- agg = xw * dis^2 + b (also serves as zero-init) -----------

__global__ void k_init_agg(const float* __restrict__ xw,
                           const float* __restrict__ dis,
                           const float* __restrict__ b,
                           float* __restrict__ agg) {
    int i = blockIdx.x * blockDim.x + threadIdx.x;   // over N*128
    int node = i >> 7, f = i & 127;
    float ds = dis[node];
    agg[i] = xw[i] * ds * ds + b[f];
}

// ---------------- edge scatter: one wave per edge ---------------------------

__global__ void k_edge_scatter(const float* __restrict__ xw,
                               const int* __restrict__ src,
                               const int* __restrict__ dst,
                               const float* __restrict__ dis,
                               float* __restrict__ agg, int e) {
    int gt = blockIdx.x * blockDim.x + threadIdx.x;
    int edge = gt >> 5;
    if (edge >= e) return;
    int lane = threadIdx.x & 31;
    int s = src[edge], d = dst[edge];
    float norm = dis[s] * dis[d];
    const float4* xs = (const float4*)(xw + (size_t)s * GD);
    float4 v = xs[lane];
    float* ad = agg + (size_t)d * GD + lane * 4;
    atomicAdd(ad + 0, v.x * norm);
    atomicAdd(ad + 1, v.y * norm);
    atomicAdd(ad + 2, v.z * norm);
    atomicAdd(ad + 3, v.w * norm);
}

// ---------------- h = relu(agg [+ h]) ---------------------------------------

__global__ void k_finish(const float* __restrict__ agg,
                         float* __restrict__ h, int residual) {
    int i = blockIdx.x * blockDim.x + threadIdx.x;   // over N*128
    float v = agg[i] + (residual ? h[i] : 0.0f);
    h[i] = fmaxf(v, 0.0f);
}

// ---------------- output layer: D=1 -----------------------------------------

__global__ void k_dot(const float* __restrict__ h,
                      const float* __restrict__ Wout,
                      float* __restrict__ xw1, int n) {
    int gt = blockIdx.x * blockDim.x + threadIdx.x;
    int row = gt >> 5;
    if (row >= n) return;
    int lane = threadIdx.x & 31;
    float4 a = ((const float4*)(h + (size_t)row * GD))[lane];
    float4 b = ((const float4*)Wout)[lane];
    float sum = a.x * b.x + a.y * b.y + a.z * b.z + a.w * b.w;
    #pragma unroll
    for (int off = 16; off > 0; off >>= 1)
        sum += __shfl_down(sum, off, 32);
    if (lane == 0) xw1[row] = sum;
}

__global__ void k_out_init(const float* __restrict__ xw1,
                           const float* __restrict__ dis,
                           const float* __restrict__ bout,
                           float* __restrict__ out, int n) {
    int i = blockIdx.x * blockDim.x + threadIdx.x;
    if (i < n) {
        float ds = dis[i];
        out[i] = xw1[i] * ds * ds + bout[0];
    }
}

__global__ void k_out_scatter(const float* __restrict__ xw1,
                              const int* __restrict__ src,
                              const int* __restrict__ dst,
                              const float* __restrict__ dis,
                              float* __restrict__ out, int e) {
    int i = blockIdx.x * blockDim.x + threadIdx.x;
    if (i < e) {
        int s = src[i], d = dst[i];
        atomicAdd(&out[d], xw1[s] * dis[s] * dis[d]);
    }
}

// ---------------- launch ----------------------------------------------------

extern "C" void kernel_launch(void* const* d_in, const int* in_sizes, int n_in,
                              void* d_out, int out_size, void* d_ws, size_t ws_size,
                              hipStream_t stream) {
    (void)in_sizes; (void)n_in; (void)out_size; (void)ws_size;
    const float* x     = (const float*)d_in[0];
    const int*   eidx  = (const int*)d_in[1];
    const float* W_in  = (const float*)d_in[2];
    const float* b_in  = (const float*)d_in[3];
    const float* W_h   = (const float*)d_in[4];   // [NLAYER,128,128]
    const float* b_h   = (const float*)d_in[5];   // [NLAYER,128]
    const float* W_out = (const float*)d_in[6];   // [128,1]
    const float* b_out = (const float*)d_in[7];
    float* out = (float*)d_out;

    const int* src = eidx;
    const int* dst = eidx + GE;

    // workspace layout (floats)
    const size_t NPAD = (GN + 63) & ~(size_t)63;
    float* dis = (float*)d_ws;
    float* xw1 = dis + NPAD;
    float* h   = xw1 + NPAD;
    float* xw  = h + (size_t)GN * GD;
    float* agg = xw + (size_t)GN * GD;

    const int T = 256;
    const int gN    = (GN + T - 1) / T;
    const int gE    = (GE + T - 1) / T;
    const int gNF   = (GN * GD) / T;          // N*128 divisible by 256
    const int gEW   = (GE * 32) / T;          // wave per edge
    const int gGemm = GN / 16;                // 3125

    // 1) deg -> dis
    k_zero<<<gN, T, 0, stream>>>(dis, GN);
    k_deg<<<gE, T, 0, stream>>>(dst, dis, GE);
    k_rsqrt<<<gN, T, 0, stream>>>(dis, GN);

    // 2) input layer
    k_gemm128<<<gGemm, T, 0, stream>>>(x, W_in, xw);
    k_init_agg<<<gNF, T, 0, stream>>>(xw, dis, b_in, agg);
    k_edge_scatter<<<gEW, T, 0, stream>>>(xw, src, dst, dis, agg, GE);
    k_finish<<<gNF, T, 0, stream>>>(agg, h, 0);

    // 3) hidden layers with residual
    for (int l = 0; l < NLAYER; ++l) {
        const float* Wl = W_h + (size_t)l * GD * GD;
        const float* bl = b_h + (size_t)l * GD;
        k_gemm128<<<gGemm, T, 0, stream>>>(h, Wl, xw);
        k_init_agg<<<gNF, T, 0, stream>>>(xw, dis, bl, agg);
        k_edge_scatter<<<gEW, T, 0, stream>>>(xw, src, dst, dis, agg, GE);
        k_finish<<<gNF, T, 0, stream>>>(agg, h, 1);
    }

    // 4) output layer (D=1)
    k_dot<<<(GN * 32) / T, T, 0, stream>>>(h, W_out, xw1, GN);
    k_out_init<<<gN, T, 0, stream>>>(xw1, dis, b_out, out, GN);
    k_out_scatter<<<gE, T, 0, stream>>>(xw1, src, dst, dis, out, GE);
}